// CWCT_46643344835249
// MI455X (gfx1250) — compile-verified
//
#include <hip/hip_runtime.h>
#include <hip/hip_bf16.h>

// ---------------------------------------------------------------------------
// Per-label whitening-coloring transform (CWCT), CDNA5 / gfx1250, wave32.
// FP32 WMMA (v_wmma_f32_16x16x4_f32) for the covariance and apply GEMMs.
// Deterministic throughout: no float atomics, stable ballot-based compaction,
// fixed-order LDS reductions.
// ---------------------------------------------------------------------------

#define BB 2
#define CC 256
#define HH 256
#define WW 256
#define NN (HH * WW)     // 65536 pixels
#define NL 8             // labels

typedef float v2f __attribute__((ext_vector_type(2)));
typedef float v8f __attribute__((ext_vector_type(8)));

__device__ __forceinline__ v8f wmma_f32_16x16x4(v2f a, v2f b, v8f c) {
  // (neg_a, A, neg_b, B, c_mod, C, reuse_a, reuse_b)
  return __builtin_amdgcn_wmma_f32_16x16x4_f32(false, a, false, b, (short)0, c,
                                               false, false);
}

// ---------------------------------------------------------------------------
// 1) Per-block label histogram (content mask y=0, style mask y=1)
// ---------------------------------------------------------------------------
__global__ void cwct_hist(const long long* __restrict__ cm,
                          const long long* __restrict__ sm,
                          int* __restrict__ blockcounts) {
  __shared__ int hc[NL];
  const int tid = threadIdx.x;
  const int m = blockIdx.y;
  const long long* msk = m ? sm : cm;
  if (tid < NL) hc[tid] = 0;
  __syncthreads();
  const int p = blockIdx.x * 256 + tid;
  atomicAdd(&hc[(int)msk[p] & 7], 1);
  __syncthreads();
  if (tid < NL) blockcounts[(m * 256 + blockIdx.x) * NL + tid] = hc[tid];
}

// ---------------------------------------------------------------------------
// 2) Scan: label totals, label base offsets, per-block bases, guide flags
// ---------------------------------------------------------------------------
__global__ void cwct_scan(const int* __restrict__ blockcounts,
                          int* __restrict__ blockbase,
                          int* __restrict__ counts,
                          int* __restrict__ offsets,
                          int* __restrict__ guide) {
  const int l = threadIdx.x;  // 8 threads
  for (int m = 0; m < 2; ++m) {
    int tot = 0;
    for (int blk = 0; blk < 256; ++blk) tot += blockcounts[(m * 256 + blk) * NL + l];
    counts[m * NL + l] = tot;
  }
  __syncthreads();
  for (int m = 0; m < 2; ++m) {
    int base = 0;
    for (int l2 = 0; l2 < l; ++l2) base += counts[m * NL + l2];
    offsets[m * NL + l] = base;
    int run = base;
    for (int blk = 0; blk < 256; ++blk) {
      blockbase[(m * 256 + blk) * NL + l] = run;
      run += blockcounts[(m * 256 + blk) * NL + l];
    }
  }
  const int a = counts[0 * NL + l];
  const int b = counts[1 * NL + l];
  guide[l] = (a > 10) && (b > 10) && (a < 10 * b) && (b < 10 * a);
}

// ---------------------------------------------------------------------------
// 3) Stable compaction: perm[m][offset[l] + rank] = pixel  (wave32 ballots)
// ---------------------------------------------------------------------------
__global__ void cwct_scatter(const long long* __restrict__ cm,
                             const long long* __restrict__ sm,
                             const int* __restrict__ blockbase,
                             int* __restrict__ perm) {
  __shared__ int wcnt[8][NL];  // [wave][label]
  const int tid = threadIdx.x;
  const int m = blockIdx.y;
  const long long* msk = m ? sm : cm;
  const int p = blockIdx.x * 256 + tid;
  const int lbl = (int)msk[p] & 7;
  const unsigned lane = tid & 31;
  const int wave = tid >> 5;
  const unsigned lt = (lane == 0) ? 0u : ((1u << lane) - 1u);
  unsigned rank = 0;
  for (int l = 0; l < NL; ++l) {
    unsigned b32 = (unsigned)__ballot(lbl == l);
    if (l == lbl) rank = (unsigned)__popc(b32 & lt);
    if (lane == 0) wcnt[wave][l] = __popc(b32);
  }
  __syncthreads();
  int base = blockbase[(m * 256 + blockIdx.x) * NL + lbl];
  for (int w = 0; w < wave; ++w) base += wcnt[w][lbl];
  perm[m * NN + base + (int)rank] = p;
}

// ---------------------------------------------------------------------------
// 4) Masked per-label means. One block per (ch, b, feat); deterministic
//    fixed-order tree reduction in LDS.
//    means[((feat*B + b)*C + ch)*NL + l]
// ---------------------------------------------------------------------------
__global__ void cwct_means(const float* __restrict__ cf,
                           const float* __restrict__ sf,
                           const long long* __restrict__ cm,
                           const long long* __restrict__ sm,
                           const int* __restrict__ counts,
                           float* __restrict__ means) {
  __shared__ float red[256];
  const int tid = threadIdx.x;
  const int ch = blockIdx.x;
  const int b = blockIdx.y;
  const int feat = blockIdx.z;
  const float* src = feat ? sf : cf;
  const long long* msk = feat ? sm : cm;
  const size_t base = ((size_t)(b * CC + ch)) * NN;
  float acc[NL];
#pragma unroll
  for (int l = 0; l < NL; ++l) acc[l] = 0.f;
  for (int p = tid; p < NN; p += 256) acc[(int)msk[p] & 7] += src[base + p];
  for (int l = 0; l < NL; ++l) {
    red[tid] = acc[l];
    __syncthreads();
    for (int s = 128; s > 0; s >>= 1) {
      if (tid < s) red[tid] += red[tid + s];
      __syncthreads();
    }
    if (tid == 0) {
      const float cnt = (float)counts[feat * NL + l];
      means[((size_t)(feat * BB + b) * CC + ch) * NL + l] = red[0] / cnt;
    }
    __syncthreads();
  }
}

// ---------------------------------------------------------------------------
// 5) Covariance GEMM: cov = (A - mu)(A - mu)^T / (n-1) over gathered columns.
//    Grid: (4,4, feat+2*(b*8+lbl)) -> 64x64 output tile of the 256x256 cov.
//    K-panels of 64 gathered pixels staged in LDS; v_wmma_f32_16x16x4_f32.
// ---------------------------------------------------------------------------
__global__ void cwct_cov(const float* __restrict__ cf,
                         const float* __restrict__ sf,
                         const int* __restrict__ perm,
                         const int* __restrict__ counts,
                         const int* __restrict__ offsets,
                         const int* __restrict__ guide,
                         const float* __restrict__ means,
                         float* __restrict__ cov_c,
                         float* __restrict__ cov_s) {
  const int z = blockIdx.z;
  const int feat = z & 1;
  const int idx = z >> 1;
  const int b = idx >> 3;
  const int lbl = idx & 7;
  if (!guide[lbl]) return;  // block-uniform exit

  __shared__ float AM[64][65];
  __shared__ float AN[64][65];

  const int n = counts[feat * NL + lbl];
  const int off = offsets[feat * NL + lbl];
  const int* pm = perm + feat * NN;
  const float* src = feat ? sf : cf;
  float* cov = feat ? cov_s : cov_c;

  const int tid = threadIdx.x;
  const int Mb = blockIdx.y * 64;
  const int Nb = blockIdx.x * 64;

  const unsigned lane = tid & 31u;
  const int wave = tid >> 5;
  const int m0 = (wave & 3) * 16;
  const int n0 = (wave >> 2) * 32;
  const int lm = (int)(lane & 15u);
  const int k0 = (int)(lane >> 4u) * 2;

  v8f acc0 = {};
  v8f acc1 = {};

  for (int kc = 0; kc < n; kc += 64) {
    // stage gathered, centered K-panels (zeros past n -> contribute nothing)
#pragma unroll
    for (int e = 0; e < 16; ++e) {
      const int lin = e * 256 + tid;
      const int ch = lin >> 6;
      const int px = lin & 63;
      const int g = kc + px;
      float vm = 0.f, vn = 0.f;
      if (g < n) {
        const int pix = pm[off + g];
        const int chM = Mb + ch;
        const int chN = Nb + ch;
        vm = src[((size_t)(b * CC + chM)) * NN + pix] -
             means[((size_t)(feat * BB + b) * CC + chM) * NL + lbl];
        vn = src[((size_t)(b * CC + chN)) * NN + pix] -
             means[((size_t)(feat * BB + b) * CC + chN) * NL + lbl];
      }
      AM[ch][px] = vm;
      AN[ch][px] = vn;
    }
    __syncthreads();
#pragma unroll
    for (int kk = 0; kk < 64; kk += 4) {
      v2f a, b0, b1;
      a.x = AM[m0 + lm][kk + k0];
      a.y = AM[m0 + lm][kk + k0 + 1];
      b0.x = AN[n0 + lm][kk + k0];
      b0.y = AN[n0 + lm][kk + k0 + 1];
      b1.x = AN[n0 + 16 + lm][kk + k0];
      b1.y = AN[n0 + 16 + lm][kk + k0 + 1];
      acc0 = wmma_f32_16x16x4(a, b0, acc0);
      acc1 = wmma_f32_16x16x4(a, b1, acc1);
    }
    __syncthreads();
  }

  const float scale = 1.0f / ((float)n - 1.0f);
  float* dst = cov + (size_t)(b * NL + lbl) * (CC * CC);
  const int mrow = Mb + m0 + ((int)(lane >> 4u)) * 8;
#pragma unroll
  for (int r = 0; r < 8; ++r) {
    const int mg = mrow + r;
    dst[(size_t)mg * CC + (Nb + n0 + lm)] = acc0[r] * scale;
    dst[(size_t)mg * CC + (Nb + n0 + 16 + lm)] = acc1[r] * scale;
  }
}

// ---------------------------------------------------------------------------
// 6) In-place Cholesky (lower), one workgroup (256 threads) per matrix.
//    Column k staged in LDS; column j owned by thread j for the update.
// ---------------------------------------------------------------------------
__global__ void cwct_chol(float* __restrict__ cov_c, float* __restrict__ cov_s,
                          const int* __restrict__ guide) {
  const int z = blockIdx.x;
  const int feat = z & 1;
  const int idx = z >> 1;
  const int lbl = idx & 7;
  if (!guide[lbl]) return;
  float* A = (feat ? cov_s : cov_c) + (size_t)idx * (CC * CC);

  __shared__ float lcol[CC];
  __shared__ float dsh;
  const int tid = threadIdx.x;

  for (int k = 0; k < CC; ++k) {
    if (tid == 0) dsh = sqrtf(A[(size_t)k * CC + k]);
    __syncthreads();
    const float d = dsh;
    float v = 0.f;
    if (tid >= k) {
      v = (tid == k) ? d : (A[(size_t)tid * CC + k] / d);
      A[(size_t)tid * CC + k] = v;
    }
    lcol[tid] = v;
    __threadfence();
    __syncthreads();
    if (tid > k) {
      const float ljk = lcol[tid];
      for (int i = tid; i < CC; ++i)
        A[(size_t)i * CC + tid] -= lcol[i] * ljk;
    }
    __threadfence();
    __syncthreads();
  }
}

// ---------------------------------------------------------------------------
// 7) Triangular inverse of content Cholesky: Lc = inv(L). Column per thread.
// ---------------------------------------------------------------------------
__global__ void cwct_trinv(const float* __restrict__ cov_c,
                           float* __restrict__ Lc,
                           const int* __restrict__ guide) {
  const int idx = blockIdx.x;  // b*8+lbl
  const int lbl = idx & 7;
  if (!guide[lbl]) return;
  const float* Lm = cov_c + (size_t)idx * (CC * CC);
  float* X = Lc + (size_t)idx * (CC * CC);
  const int j = threadIdx.x;
  for (int i = 0; i < j; ++i) X[(size_t)i * CC + j] = 0.f;
  X[(size_t)j * CC + j] = 1.0f / Lm[(size_t)j * CC + j];
  for (int i = j + 1; i < CC; ++i) {
    float s = 0.f;
    for (int t = j; t < i; ++t)
      s += Lm[(size_t)i * CC + t] * X[(size_t)t * CC + j];
    X[(size_t)i * CC + j] = -s / Lm[(size_t)i * CC + i];
  }
}

// ---------------------------------------------------------------------------
// 8) T = Ls * Lc  (lower * lower -> lower; zero-fill upper)
// ---------------------------------------------------------------------------
__global__ void cwct_tmat(const float* __restrict__ cov_s,  // holds Ls in lower
                          const float* __restrict__ Lc,
                          float* __restrict__ Tm,
                          const int* __restrict__ guide) {
  const int idx = blockIdx.z;  // b*8+lbl
  const int lbl = idx & 7;
  if (!guide[lbl]) return;
  const float* Ls = cov_s + (size_t)idx * (CC * CC);
  const float* Li = Lc + (size_t)idx * (CC * CC);
  float* T = Tm + (size_t)idx * (CC * CC);
  const int i = blockIdx.y * 16 + threadIdx.y;
  const int j = blockIdx.x * 16 + threadIdx.x;
  float s = 0.f;
  if (i >= j)
    for (int k = j; k <= i; ++k)
      s += Ls[(size_t)i * CC + k] * Li[(size_t)k * CC + j];
  T[(size_t)i * CC + j] = s;
}

// ---------------------------------------------------------------------------
// 9) out = cf (float4 streaming copy)
// ---------------------------------------------------------------------------
__global__ void cwct_copy(const float4* __restrict__ in, float4* __restrict__ out) {
  const size_t i = (size_t)blockIdx.x * 256 + threadIdx.x;
  out[i] = in[i];
}

// ---------------------------------------------------------------------------
// 10) Apply: colored = T * (cf - mu_c) + mu_s on guided-label columns,
//     scattered through the content perm list. 256x64 output tiles,
//     K = 256 (channels) in 16-wide LDS panels, wmma f32 16x16x4.
// ---------------------------------------------------------------------------
__global__ void cwct_apply(const float* __restrict__ cf,
                           const int* __restrict__ perm,
                           const int* __restrict__ counts,
                           const int* __restrict__ offsets,
                           const int* __restrict__ guide,
                           const float* __restrict__ means,
                           const float* __restrict__ Tm,
                           float* __restrict__ out) {
  const int lbl = blockIdx.y;
  const int b = blockIdx.z;
  if (!guide[lbl]) return;
  const int n = counts[0 * NL + lbl];
  const int col0 = blockIdx.x * 64;
  if (col0 >= n) return;  // block-uniform
  const int off = offsets[0 * NL + lbl];

  __shared__ float Ap[256][17];  // T panel  [M=256][K=16]
  __shared__ float Bp[16][65];   // centered gathered columns [K=16][N=64]

  const int tid = threadIdx.x;
  const unsigned lane = tid & 31u;
  const int wave = tid >> 5;
  const int lm = (int)(lane & 15u);
  const int k0 = (int)(lane >> 4u) * 2;
  const float* T = Tm + (size_t)(b * NL + lbl) * (CC * CC);

  v8f acc[8];
#pragma unroll
  for (int q = 0; q < 8; ++q) acc[q] = (v8f){};

  for (int kc = 0; kc < CC; kc += 16) {
    // stage T panel (rows 0..255, k-chunk of 16)
#pragma unroll
    for (int e = 0; e < 16; ++e) {
      const int lin = e * 256 + tid;
      const int m = lin >> 4;
      const int kk = lin & 15;
      Ap[m][kk] = T[(size_t)m * CC + (kc + kk)];
    }
    // stage centered gathered B panel
#pragma unroll
    for (int e = 0; e < 4; ++e) {
      const int lin = e * 256 + tid;
      const int kk = lin >> 6;
      const int col = lin & 63;
      const int cg = col0 + col;
      float v = 0.f;
      if (cg < n) {
        const int pix = perm[0 * NN + off + cg];
        const int ch = kc + kk;
        v = cf[((size_t)(b * CC + ch)) * NN + pix] -
            means[((size_t)(0 * BB + b) * CC + ch) * NL + lbl];
      }
      Bp[kk][col] = v;
    }
    __syncthreads();
#pragma unroll
    for (int ks = 0; ks < 16; ks += 4) {
      v2f a0, a1;
      const int mr0 = wave * 32;
      a0.x = Ap[mr0 + lm][ks + k0];
      a0.y = Ap[mr0 + lm][ks + k0 + 1];
      a1.x = Ap[mr0 + 16 + lm][ks + k0];
      a1.y = Ap[mr0 + 16 + lm][ks + k0 + 1];
#pragma unroll
      for (int nt = 0; nt < 4; ++nt) {
        v2f bf;
        bf.x = Bp[ks + k0][nt * 16 + lm];
        bf.y = Bp[ks + k0 + 1][nt * 16 + lm];
        acc[nt] = wmma_f32_16x16x4(a0, bf, acc[nt]);
        acc[4 + nt] = wmma_f32_16x16x4(a1, bf, acc[4 + nt]);
      }
    }
    __syncthreads();
  }

  // add style mean and scatter to out
  const int mbase = wave * 32 + ((int)(lane >> 4u)) * 8;
#pragma unroll
  for (int t = 0; t < 2; ++t) {
#pragma unroll
    for (int nt = 0; nt < 4; ++nt) {
      const int ng = nt * 16 + lm;
      const int cg = col0 + ng;
      if (cg < n) {
        const int pix = perm[0 * NN + off + cg];
#pragma unroll
        for (int r = 0; r < 8; ++r) {
          const int mg = mbase + t * 16 + r;
          const float sm =
              means[((size_t)(1 * BB + b) * CC + mg) * NL + lbl];
          out[((size_t)(b * CC + mg)) * NN + pix] = acc[t * 4 + nt][r] + sm;
        }
      }
    }
  }
}

// ---------------------------------------------------------------------------
// Launch
// ---------------------------------------------------------------------------
extern "C" void kernel_launch(void* const* d_in, const int* in_sizes, int n_in,
                              void* d_out, int out_size, void* d_ws, size_t ws_size,
                              hipStream_t stream) {
  const float* cf = (const float*)d_in[0];
  const float* sf = (const float*)d_in[1];
  const long long* cm = (const long long*)d_in[2];
  const long long* sm = (const long long*)d_in[3];
  float* out = (float*)d_out;

  // workspace carve (~17.4 MB total)
  int* blockcounts = (int*)d_ws;                 // 2*256*8
  int* blockbase = blockcounts + 4096;           // 2*256*8
  int* counts = blockbase + 4096;                // 2*8
  int* offsets = counts + 16;                    // 2*8
  int* guide = offsets + 16;                     // 8
  int* perm = guide + 8;                         // 2*65536
  float* means = (float*)(perm + 2 * NN);        // 2*2*256*8
  float* cov_c = means + 2 * BB * CC * NL;       // 16*65536
  float* cov_s = cov_c + (size_t)BB * NL * CC * CC;
  float* Lc = cov_s + (size_t)BB * NL * CC * CC;
  float* Tm = Lc + (size_t)BB * NL * CC * CC;

  cwct_hist<<<dim3(256, 2), 256, 0, stream>>>(cm, sm, blockcounts);
  cwct_scan<<<1, 8, 0, stream>>>(blockcounts, blockbase, counts, offsets, guide);
  cwct_scatter<<<dim3(256, 2), 256, 0, stream>>>(cm, sm, blockbase, perm);
  cwct_means<<<dim3(CC, BB, 2), 256, 0, stream>>>(cf, sf, cm, sm, counts, means);
  cwct_cov<<<dim3(4, 4, 2 * BB * NL), 256, 0, stream>>>(
      cf, sf, perm, counts, offsets, guide, means, cov_c, cov_s);
  cwct_chol<<<2 * BB * NL, 256, 0, stream>>>(cov_c, cov_s, guide);
  cwct_trinv<<<BB * NL, 256, 0, stream>>>(cov_c, Lc, guide);
  cwct_tmat<<<dim3(16, 16, BB * NL), dim3(16, 16), 0, stream>>>(cov_s, Lc, Tm, guide);
  cwct_copy<<<(BB * CC * NN) / (256 * 4), 256, 0, stream>>>((const float4*)cf,
                                                            (float4*)out);
  cwct_apply<<<dim3(NN / 64, NL, BB), 256, 0, stream>>>(
      cf, perm, counts, offsets, guide, means, Tm, out);
}